// LinearSelfAttention_29042568856115
// MI455X (gfx1250) — compile-verified
//
#include <hip/hip_runtime.h>
#include <cstddef>

#define CDIM 512
#define PDIM 64
#define NDIM 256
#define CSTRIDE 16384   // PDIM*NDIM, stride between channels in x/out
#define KC 64           // K-chunk staged in LDS for GEMM1
#define NT 64           // N columns per workgroup in kernel 2
#define PRS 130         // floats per k-pair row: 64 cols * 2 interleaved + 2 pad (even -> 8B align)

typedef float v2f __attribute__((ext_vector_type(2)));
typedef float v8f __attribute__((ext_vector_type(8)));

// ---------------------------------------------------------------------------
// Kernel 1: per (b,p) -> softmax(q) collapsed with K-GEMM into cv[512]
//   q[n]  = w_qkv[0,:] . x[b,:,p,n] + b_qkv[0]
//   s     = softmax(q)
//   t[c]  = x[b,c,p,:] . s
//   cv[o] = w_qkv[1+o,:] . t + b_qkv[1+o]        (valid since sum(s)==1)
// ---------------------------------------------------------------------------
__global__ void __launch_bounds__(256) lsa_k1_cv(
    const float* __restrict__ x, const float* __restrict__ wqkv,
    const float* __restrict__ bqkv, float* __restrict__ cv_ws)
{
    __shared__ float w0s[CDIM];
    __shared__ float red[256];
    __shared__ float scores[NDIM];
    __shared__ float ts[CDIM];

    const int tid = threadIdx.x;
    const int bp  = blockIdx.x;
    const int b   = bp >> 6;
    const int p   = bp & 63;
    const float* xb = x + (size_t)b * CDIM * CSTRIDE + (size_t)p * NDIM;

    w0s[tid]       = wqkv[tid];
    w0s[tid + 256] = wqkv[tid + 256];
    __syncthreads();

    // q for n = tid (coalesced across threads)
    float q = bqkv[0];
    for (int c = 0; c < CDIM; ++c)
        q += w0s[c] * xb[(size_t)c * CSTRIDE + tid];

    // softmax over the 256 n-values
    red[tid] = q; __syncthreads();
    for (int s = 128; s > 0; s >>= 1) {
        if (tid < s) red[tid] = fmaxf(red[tid], red[tid + s]);
        __syncthreads();
    }
    const float m = red[0]; __syncthreads();
    const float e = __expf(q - m);
    red[tid] = e; __syncthreads();
    for (int s = 128; s > 0; s >>= 1) {
        if (tid < s) red[tid] += red[tid + s];
        __syncthreads();
    }
    scores[tid] = e * (1.0f / red[0]);
    __syncthreads();

    // t[c] = x-row(c) . scores   (each thread streams 2 contiguous 1KB rows)
    for (int c = tid; c < CDIM; c += 256) {
        const float* xr = xb + (size_t)c * CSTRIDE;
        float acc = 0.0f;
        for (int n = 0; n < NDIM; ++n) acc += xr[n] * scores[n];
        ts[c] = acc;
    }
    __syncthreads();

    // cv[o] = W_k-row(o) . t + b_k[o]
    for (int o = tid; o < CDIM; o += 256) {
        const float* wr = wqkv + (size_t)(1 + o) * CDIM;
        float acc = bqkv[1 + o];
        for (int c = 0; c < CDIM; ++c) acc += wr[c] * ts[c];
        cv_ws[(size_t)bp * CDIM + o] = acc;
    }
}

// ---------------------------------------------------------------------------
// Kernel 2: per (b, p, 64-col tile), fully fused in LDS:
//   V' = relu(W_v @ X + b_v) * cv           (WMMA GEMM 1, result -> LDS)
//   out = W_out @ V' + b_out                (WMMA GEMM 2, B-frags from LDS)
// 256 threads = 8 wave32; wave w owns rows [64w, 64w+64) x all 64 cols:
// 4x4 tiles of 16x16 -> v8f acc[4][4] per wave.
// LDS tiles are stored k-pair interleaved ([k/2][n][2]) so every WMMA
// B-fragment {B[k][n], B[k+1][n]} is one aligned ds_load_b64 into a VGPR
// pair; X staging is register double-buffered against the WMMA loop.
// ---------------------------------------------------------------------------
__global__ void __launch_bounds__(256) lsa_k2_fused(
    const float* __restrict__ x, const float* __restrict__ wqkv,
    const float* __restrict__ bqkv, const float* __restrict__ wout,
    const float* __restrict__ bout, const float* __restrict__ cv_ws,
    float* __restrict__ out)
{
    extern __shared__ float lds[];
    float* xs  = lds;                        // (KC/2)*PRS      X chunk (pairs)
    float* vs  = xs  + (KC / 2) * PRS;       // (CDIM/2)*PRS    V' tile (pairs)
    float* cvs = vs  + (CDIM / 2) * PRS;     // CDIM
    float* bvs = cvs + CDIM;                 // CDIM (value bias)
    float* bos = bvs + CDIM;                 // CDIM (out bias)

    const int tid  = threadIdx.x;
    const int lane = tid & 31;
    const int wv   = tid >> 5;           // wave 0..7
    const int bp   = blockIdx.x >> 2;
    const int nt   = (blockIdx.x & 3) * NT;
    const int b    = bp >> 6;
    const int p    = bp & 63;

    const float* xb = x + (size_t)b * CDIM * CSTRIDE + (size_t)p * NDIM + nt;

    // preload cv + biases (value-bias rows are wqkv rows 513..1024)
    {
        int o = tid;
        cvs[o] = cv_ws[(size_t)bp * CDIM + o]; bvs[o] = bqkv[513 + o]; bos[o] = bout[o];
        o += 256;
        cvs[o] = cv_ws[(size_t)bp * CDIM + o]; bvs[o] = bqkv[513 + o]; bos[o] = bout[o];
    }

    const int mrow  = lane & 15;             // A-frag: M within tile
    const int khalf = (lane >> 4) << 1;      // A/B-frag: K pair select (0 or 2)
    const int nlane = lane & 15;             // B/C-frag: N within tile
    const int rowhi = (lane >> 4) << 3;      // C/D-frag: +8 row for hi lanes
    const int rw    = wv * 64;               // this wave's row block

    v8f acc[4][4];
#pragma unroll
    for (int i = 0; i < 4; ++i)
#pragma unroll
        for (int j = 0; j < 4; ++j) acc[i][j] = (v8f){0,0,0,0,0,0,0,0};

    // Each thread stages 8 k-pairs (one per 256-thread sweep of 2048 pairs).
    v2f st[8];
#define LOAD_CHUNK(K0)                                                        \
    {                                                                         \
        _Pragma("unroll")                                                     \
        for (int i = 0; i < 8; ++i) {                                         \
            const int idx = tid + i * 256;                                    \
            const int pr = idx >> 6, cc = idx & 63;                           \
            const float* g = xb + (size_t)((K0) + 2 * pr) * CSTRIDE + cc;     \
            st[i].x = g[0];                                                   \
            st[i].y = g[CSTRIDE];                                             \
        }                                                                     \
    }
#define STORE_CHUNK()                                                         \
    {                                                                         \
        _Pragma("unroll")                                                     \
        for (int i = 0; i < 8; ++i) {                                         \
            const int idx = tid + i * 256;                                    \
            const int pr = idx >> 6, cc = idx & 63;                           \
            *(v2f*)&xs[pr * PRS + (cc << 1)] = st[i];                         \
        }                                                                     \
    }

    // ---------------- GEMM 1: V' = relu(W_v @ X + b_v) * cv ----------------
    const float* wvbase = wqkv + (size_t)(1 + CDIM) * CDIM;   // value rows
    LOAD_CHUNK(0);
    STORE_CHUNK();
    __syncthreads();

    for (int c = 0; c < CDIM / KC; ++c) {
        const int k0 = c * KC;
        if (c + 1 < CDIM / KC) LOAD_CHUNK(k0 + KC);   // in flight during WMMA

#pragma unroll 4
        for (int kk = 0; kk < KC; kk += 4) {
            const int kg = k0 + kk + khalf;
            v2f a[4];
#pragma unroll
            for (int i = 0; i < 4; ++i)
                a[i] = *(const v2f*)(wvbase + (size_t)(rw + i * 16 + mrow) * CDIM + kg);
            v2f bbv[4];
            const int kpr = (kk + khalf) >> 1;        // pair row in xs
#pragma unroll
            for (int j = 0; j < 4; ++j)
                bbv[j] = *(const v2f*)&xs[kpr * PRS + ((j * 16 + nlane) << 1)];
#pragma unroll
            for (int i = 0; i < 4; ++i)
#pragma unroll
                for (int j = 0; j < 4; ++j)
                    acc[i][j] = __builtin_amdgcn_wmma_f32_16x16x4_f32(
                        false, a[i], false, bbv[j], (short)0, acc[i][j], false, false);
        }
        __syncthreads();
        if (c + 1 < CDIM / KC) {
            STORE_CHUNK();
            __syncthreads();
        }
    }

    // epilogue 1: relu(+bias)*cv -> pair-interleaved LDS V' (ds_store_b64)
#pragma unroll
    for (int i = 0; i < 4; ++i)
#pragma unroll
        for (int j = 0; j < 4; ++j)
#pragma unroll
            for (int r = 0; r < 8; r += 2) {
                const int row = rw + i * 16 + rowhi + r;        // even
                v2f pv;
                pv.x = fmaxf(acc[i][j][r]     + bvs[row],     0.0f) * cvs[row];
                pv.y = fmaxf(acc[i][j][r + 1] + bvs[row + 1], 0.0f) * cvs[row + 1];
                *(v2f*)&vs[(row >> 1) * PRS + ((j * 16 + nlane) << 1)] = pv;
            }
    __syncthreads();

    // ---------------- GEMM 2: out = W_out @ V' + b_out ----------------
#pragma unroll
    for (int i = 0; i < 4; ++i)
#pragma unroll
        for (int j = 0; j < 4; ++j) acc[i][j] = (v8f){0,0,0,0,0,0,0,0};

#pragma unroll 4
    for (int k = 0; k < CDIM; k += 4) {
        const int kg = k + khalf;
        v2f a[4];
#pragma unroll
        for (int i = 0; i < 4; ++i)
            a[i] = *(const v2f*)(wout + (size_t)(rw + i * 16 + mrow) * CDIM + kg);
        v2f bbv[4];
        const int kpr = kg >> 1;                      // pair row in vs
#pragma unroll
        for (int j = 0; j < 4; ++j)
            bbv[j] = *(const v2f*)&vs[kpr * PRS + ((j * 16 + nlane) << 1)];
#pragma unroll
        for (int i = 0; i < 4; ++i)
#pragma unroll
            for (int j = 0; j < 4; ++j)
                acc[i][j] = __builtin_amdgcn_wmma_f32_16x16x4_f32(
                    false, a[i], false, bbv[j], (short)0, acc[i][j], false, false);
    }

    // epilogue 2: +b_out, coalesced stores
    float* ob = out + (size_t)b * CDIM * CSTRIDE + (size_t)p * NDIM + nt;
#pragma unroll
    for (int i = 0; i < 4; ++i)
#pragma unroll
        for (int j = 0; j < 4; ++j)
#pragma unroll
            for (int r = 0; r < 8; ++r) {
                const int row = rw + i * 16 + rowhi + r;
                ob[(size_t)row * CSTRIDE + j * 16 + nlane] = acc[i][j][r] + bos[row];
            }
#undef LOAD_CHUNK
#undef STORE_CHUNK
}

// ---------------------------------------------------------------------------
extern "C" void kernel_launch(void* const* d_in, const int* in_sizes, int n_in,
                              void* d_out, int out_size, void* d_ws, size_t ws_size,
                              hipStream_t stream) {
    const float* x     = (const float*)d_in[0];
    const float* wqkv  = (const float*)d_in[1];
    const float* bqkv  = (const float*)d_in[2];
    const float* wout  = (const float*)d_in[3];
    const float* bout  = (const float*)d_in[4];
    float* outp  = (float*)d_out;
    float* cv_ws = (float*)d_ws;                     // 1024 * 512 floats = 2 MB

    const int BP = 16 * PDIM;                        // 1024 (b,p) pairs

    lsa_k1_cv<<<BP, 256, 0, stream>>>(x, wqkv, bqkv, cv_ws);

    const size_t lds_bytes =
        (size_t)((KC / 2) * PRS + (CDIM / 2) * PRS + 3 * CDIM) * sizeof(float); // ~152 KB
    lsa_k2_fused<<<BP * (NDIM / NT), 256, lds_bytes, stream>>>(
        x, wqkv, bqkv, wout, bout, cv_ws, outp);
}